// GraphConvolutionLayer_11158325035210
// MI455X (gfx1250) — compile-verified
//
#include <hip/hip_runtime.h>
#include <hip/hip_bf16.h>

typedef __attribute__((ext_vector_type(2))) float v2f;
typedef __attribute__((ext_vector_type(8))) float v8f;

#define D_IN  64
#define D_OUT 32

// ---------------------------------------------------------------------------
// Kernel 1: XW = X @ W^T using native fp32 WMMA (v_wmma_f32_16x16x4_f32).
// One wave computes one 16x16 tile of XW. K=64 -> 16 WMMA steps of K=4.
//
// A (16x4 f32) per-lane layout (ISA 7.12.2):
//   lanes 0-15 : M=lane,    vgpr0=A[M][k+0], vgpr1=A[M][k+1]
//   lanes 16-31: M=lane-16, vgpr0=A[M][k+2], vgpr1=A[M][k+3]
// B (4x16 f32) is symmetric with N in place of M. Since B[k][n] = W[n][k]
// (W stored [out,in]), both operands are a float2 load from a row at col
// k + 2*(lane>>4).
//
// C/D layout: vgpr v -> row = m_base + v + 8*(lane>>4), col = n_base + (lane&15).
// ---------------------------------------------------------------------------
__global__ void gcn_gemm_wmma_f32(const float* __restrict__ X,
                                  const float* __restrict__ W,
                                  float* __restrict__ XW,
                                  int n_nodes, int row_tiles) {
  const int lane  = threadIdx.x & 31;
  const int wave  = (blockIdx.x * (blockDim.x >> 5)) + (threadIdx.x >> 5);
  const int rowTile = wave >> 1;       // which 16-row tile of X
  const int colTile = wave & 1;        // which 16-col tile of XW (D_OUT=32)
  if (rowTile >= row_tiles) return;    // whole-wave uniform exit

  const int m_base = rowTile * 16;
  const int n_base = colTile * 16;
  const int half   = lane >> 4;        // 0: K pair (k,k+1), 1: (k+2,k+3)
  const int l15    = lane & 15;

  // Clamp row for safety on a ragged tail (n_nodes here is a multiple of 16).
  int arow_idx = m_base + l15;
  if (arow_idx >= n_nodes) arow_idx = n_nodes - 1;

  const float* __restrict__ arow = X + (long long)arow_idx * D_IN;
  const float* __restrict__ brow = W + (long long)(n_base + l15) * D_IN;

  v8f c = {};
#pragma unroll
  for (int k = 0; k < D_IN; k += 4) {
    const int kk = k + 2 * half;                         // even -> 8B aligned
    const v2f a = *(const v2f*)(arow + kk);
    const v2f b = *(const v2f*)(brow + kk);
    // (neg_a, A, neg_b, B, c_mod, C, reuse_a, reuse_b)
    c = __builtin_amdgcn_wmma_f32_16x16x4_f32(false, a, false, b,
                                              (short)0, c, false, false);
  }

  // Store D: vgpr v holds row m_base + v + 8*half, col n_base + l15.
  const int row0 = m_base + 8 * half;
  float* __restrict__ obase = XW + (long long)row0 * D_OUT + n_base + l15;
#pragma unroll
  for (int v = 0; v < 8; ++v) {
    if (row0 + v < n_nodes) obase[(long long)v * D_OUT] = c[v];
  }
}

// ---------------------------------------------------------------------------
// Kernel 2: zero the output (atomic accumulation target).
// ---------------------------------------------------------------------------
__global__ void gcn_zero_out(float* __restrict__ out, int n) {
  const int i = blockIdx.x * blockDim.x + threadIdx.x;
  if (i < n) out[i] = 0.0f;
}

// ---------------------------------------------------------------------------
// Kernel 3: edge scatter. One wave per edge, lane = output channel.
//  - edge index is forced wave-uniform (readfirstlane) so the three edge
//    scalars become SMEM broadcast loads instead of per-lane VMEM.
//  - the XW gather is one coalesced 128B load per wave.
//  - the accumulation uses the CDNA5 hardware f32 atomic
//    (GLOBAL_ATOMIC_ADD_F32, no-return form) via inline asm so it can never
//    fall back to a CAS retry loop. Both XW (12.8MB) and out (12.8MB) are
//    L2-resident (192MB L2), so all 51.2M atomics resolve in L2.
// ---------------------------------------------------------------------------
__global__ void gcn_edge_scatter(const float* __restrict__ XW,
                                 const int* __restrict__ esrc,
                                 const int* __restrict__ edst,
                                 const float* __restrict__ ew,
                                 float* __restrict__ out,
                                 int n_edges) {
  const long long tid = (long long)blockIdx.x * blockDim.x + threadIdx.x;
  const int ch = (int)(tid & 31);                          // channel = lane
  const int e  = __builtin_amdgcn_readfirstlane((int)(tid >> 5)); // wave-uniform
  if (e >= n_edges) return;

  const int   s = esrc[e];   // SMEM broadcast
  const int   d = edst[e];   // SMEM broadcast
  const float w = ew[e];     // SMEM broadcast

  const float v = XW[(long long)s * D_OUT + ch] * w;
  float* addr = out + (long long)d * D_OUT + ch;

  // GLOBAL_ATOMIC_ADD_F32 without return: addr in VGPR pair, data in VGPR.
  asm volatile("global_atomic_add_f32 %0, %1, off"
               :
               : "v"(addr), "v"(v)
               : "memory");
}

extern "C" void kernel_launch(void* const* d_in, const int* in_sizes, int n_in,
                              void* d_out, int out_size, void* d_ws, size_t ws_size,
                              hipStream_t stream) {
  const float* X    = (const float*)d_in[0];   // [N, 64]
  const float* W    = (const float*)d_in[1];   // [32, 64]
  const int*   esrc = (const int*)d_in[2];     // [E]
  const int*   edst = (const int*)d_in[3];     // [E]
  const float* ew   = (const float*)d_in[4];   // [E]
  float*       out  = (float*)d_out;           // [N, 32]
  float*       XW   = (float*)d_ws;            // scratch [N, 32] = 12.8 MB

  const int n_nodes = in_sizes[0] / D_IN;
  const int n_edges = in_sizes[2];

  // 1) projection via fp32 WMMA
  const int row_tiles   = (n_nodes + 15) / 16;
  const int total_waves = row_tiles * 2;             // 2 col tiles (D_OUT=32)
  const int gemm_blocks = (total_waves + 7) / 8;     // 8 waves / 256-thread block
  gcn_gemm_wmma_f32<<<gemm_blocks, 256, 0, stream>>>(X, W, XW, n_nodes, row_tiles);

  // 2) zero accumulator
  gcn_zero_out<<<(out_size + 255) / 256, 256, 0, stream>>>(out, out_size);

  // 3) scatter-add over edges (one wave per edge)
  const long long threads = (long long)n_edges * 32;
  const int scat_blocks = (int)((threads + 255) / 256);
  gcn_edge_scatter<<<scat_blocks, 256, 0, stream>>>(XW, esrc, edst, ew, out, n_edges);
}